// constrainLayer_11218454577217
// MI455X (gfx1250) — compile-verified
//
#include <hip/hip_runtime.h>
#include <hip/hip_bf16.h>
#include <math.h>

// ---------------------------------------------------------------------------
// agg[v] = sum_{e: dst=v} (h[src_e] - h[dst_e])
//        = (sum_{e: dst=v} h[src_e]) - deg(v) * h[v]
// out[v] = agg[v] / (||agg[v]||_2 + 1e-7)
//
// Phase 1: zero accumulators.
// Phase 2: one wave32 per edge; lane l handles float4 chunk l of the 128-wide
//          row: gather h[src] (L2-resident, 25.6MB << 192MB L2) and
//          global_atomic_add_f32 into out[dst]; lane 0 bumps deg[dst].
// Phase 3: one wave32 per 16 nodes; squared row-sums of (agg - deg*h) computed
//          on the matrix pipe via 32 accumulating V_WMMA_F32_16X16X4_F32 ops,
//          then coalesced normalize+store.
// ---------------------------------------------------------------------------

typedef __attribute__((ext_vector_type(2))) float v2f;
typedef __attribute__((ext_vector_type(8))) float v8f;

#define DIMS 128

__global__ void zero_kernel(float* __restrict__ out, int n_out,
                            int* __restrict__ deg, int n_deg) {
    int i = blockIdx.x * blockDim.x + threadIdx.x;
    if (i < n_out) out[i] = 0.0f;
    if (i < n_deg) deg[i] = 0;
}

__global__ void scatter_kernel(const float* __restrict__ h,
                               const int* __restrict__ src,
                               const int* __restrict__ dst,
                               float* __restrict__ agg,
                               int* __restrict__ deg,
                               int n_edges) {
    const int lane = threadIdx.x & 31;
    const int e = blockIdx.x * (blockDim.x >> 5) + (threadIdx.x >> 5);
    if (e >= n_edges) return;               // wave-uniform exit

    const int s = src[e];
    const int d = dst[e];

    // Gather one float4 of h[src] per lane: global_load_b128, L2 hit.
    const float4 v = ((const float4*)(h + (size_t)s * DIMS))[lane];

    // Scatter-add into agg[dst]: 4x global_atomic_add_f32 (no return).
    float* drow = agg + (size_t)d * DIMS + lane * 4;
    atomicAdd(drow + 0, v.x);
    atomicAdd(drow + 1, v.y);
    atomicAdd(drow + 2, v.z);
    atomicAdd(drow + 3, v.w);

    if (lane == 0) atomicAdd(deg + d, 1);
}

__global__ void finalize_kernel(const float* __restrict__ h,
                                const int* __restrict__ deg,
                                float* __restrict__ out,   // agg in, result out
                                int n_nodes) {
    const int lane = threadIdx.x & 31;
    const int wave = blockIdx.x * (blockDim.x >> 5) + (threadIdx.x >> 5);
    const int nodeBase = wave * 16;
    if (nodeBase >= n_nodes) return;        // wave-uniform: EXEC stays full

    // --- Phase A: ssq per row via WMMA f32 16x16x4, D = (A.^2) x ones + C ---
    // A-layout (16x4 f32): lanes 0-15 hold row m, K={0,1}; lanes 16-31 hold
    // row m, K={2,3}. We walk 32 K-chunks of 4 to cover 128 columns.
    const int m       = lane & 15;
    const int node    = min(nodeBase + m, n_nodes - 1);
    const int khalf   = (lane >> 4) << 1;   // 0 or 2
    const float degv  = (float)deg[node];
    const float* grow = out + (size_t)node * DIMS;
    const float* hrow = h   + (size_t)node * DIMS;

    v8f c = {};
    v2f ones; ones[0] = 1.0f; ones[1] = 1.0f;
#pragma unroll
    for (int kk = 0; kk < 32; ++kk) {
        const int col = kk * 4 + khalf;
        const float a0 = grow[col]     - degv * hrow[col];
        const float a1 = grow[col + 1] - degv * hrow[col + 1];
        v2f A; A[0] = a0 * a0; A[1] = a1 * a1;
        c = __builtin_amdgcn_wmma_f32_16x16x4_f32(
                false, A, false, ones, (short)0, c, false, false);
    }
    // C/D layout: VGPR j -> M=j (lanes 0-15), M=j+8 (lanes 16-31); every N
    // column holds the same row-sum, so lane 0 / lane 16 carry rows 0-7 / 8-15.

    // --- Phase B: coalesced normalize + store, one row per iteration -------
#pragma unroll
    for (int r = 0; r < 16; ++r) {
        const float ssq = __shfl(c[r & 7], (r >> 3) << 4, 32);
        const float inv = 1.0f / (sqrtf(ssq) + 1e-7f);
        const int nodeR = min(nodeBase + r, n_nodes - 1);
        const float dR  = (float)deg[nodeR];
        const float4 g  = ((const float4*)(out + (size_t)nodeR * DIMS))[lane];
        const float4 hv = ((const float4*)(h   + (size_t)nodeR * DIMS))[lane];
        float4 o;
        o.x = (g.x - dR * hv.x) * inv;
        o.y = (g.y - dR * hv.y) * inv;
        o.z = (g.z - dR * hv.z) * inv;
        o.w = (g.w - dR * hv.w) * inv;
        ((float4*)(out + (size_t)nodeR * DIMS))[lane] = o;
    }
}

extern "C" void kernel_launch(void* const* d_in, const int* in_sizes, int n_in,
                              void* d_out, int out_size, void* d_ws, size_t ws_size,
                              hipStream_t stream) {
    const float* h   = (const float*)d_in[0];
    // d_in[1] = r, unused by the reference computation
    const int*  src  = (const int*)d_in[2];
    const int*  dst  = (const int*)d_in[3];
    float*      out  = (float*)d_out;
    int*        deg  = (int*)d_ws;          // n_nodes ints (200 KB) of scratch

    const int n_nodes = in_sizes[0] / DIMS;
    const int n_edges = in_sizes[2];
    const int n_out   = n_nodes * DIMS;

    // Phase 1: zero agg (accumulated in d_out) and deg.
    {
        int nmax = n_out > n_nodes ? n_out : n_nodes;
        zero_kernel<<<(nmax + 255) / 256, 256, 0, stream>>>(out, n_out, deg, n_nodes);
    }

    // Phase 2: one wave per edge, 8 waves per 256-thread block.
    {
        int blocks = (n_edges + 7) / 8;
        scatter_kernel<<<blocks, 256, 0, stream>>>(h, src, dst, out, deg, n_edges);
    }

    // Phase 3: one wave per 16 nodes.
    {
        int waves  = (n_nodes + 15) / 16;
        int blocks = (waves + 7) / 8;
        finalize_kernel<<<blocks, 256, 0, stream>>>(h, deg, out, n_nodes);
    }
}